// CircuitChannel_15444702396998
// MI455X (gfx1250) — compile-verified
//
#include <hip/hip_runtime.h>

// ---------------------------------------------------------------------------
// 24-qubit state-vector circuit (4 layers RX + CZ chain, then measure q0)
// MI455X strategy:
//  * 128 MB state is L2-resident (192 MB L2) -> multi-pass costs ~L2 BW only
//  * CZ chain is diagonal: folded into stores as popcount-parity sign flips
//  * 12 RX gates fused per pass (2 passes/layer) through a 4096-amp LDS tile
//  * each 4-RX group = 16x16 complex unitary applied via
//    V_WMMA_F32_16X16X4_F32 (4 real GEMMs = 16 WMMAs per 16-group batch)
//  * tile load/store uses GLOBAL_{LOAD,STORE}_ASYNC_*_LDS_B64 (ASYNCcnt path)
// ---------------------------------------------------------------------------

typedef __attribute__((ext_vector_type(2))) float v2f;
typedef __attribute__((ext_vector_type(8))) float v8f;
typedef __attribute__((ext_vector_type(2))) int   v2i;   // async b64 payload

#define AS1 __attribute__((address_space(1)))   // global
#define AS3 __attribute__((address_space(3)))   // LDS

#define NQ      24
#define NSTATE  (1 << NQ)     // 16,777,216 amplitudes
#define TILE    4096          // amplitudes per workgroup tile (32 KB LDS)
#define TPB     256           // 8 waves (wave32)
#define NWAVE   (TPB / 32)
#define NLAYER  4

#if defined(__has_builtin)
#  if __has_builtin(__builtin_amdgcn_global_load_async_to_lds_b64)
#    define HAS_ASYNC_LOAD 1
#  else
#    define HAS_ASYNC_LOAD 0
#  endif
#  if __has_builtin(__builtin_amdgcn_global_store_async_from_lds_b64)
#    define HAS_ASYNC_STORE 1
#  else
#    define HAS_ASYNC_STORE 0
#  endif
#  if __has_builtin(__builtin_amdgcn_s_wait_asynccnt)
#    define WAIT_ASYNC0() __builtin_amdgcn_s_wait_asynccnt(0)
#  else
#    define WAIT_ASYNC0() asm volatile("s_wait_asynccnt 0x0" ::: "memory")
#  endif
#else
#  define HAS_ASYNC_LOAD 0
#  define HAS_ASYNC_STORE 0
#  define WAIT_ASYNC0()
#endif

__device__ __forceinline__ v8f wmma_f32_16x16x4(v2f a, v2f b, v8f c) {
  // D = A(16x4) * B(4x16) + C  (f32 WMMA; NEG supports only C-negate, unused)
  return __builtin_amdgcn_wmma_f32_16x16x4_f32(
      /*neg_a=*/false, a, /*neg_b=*/false, b,
      /*c_mod=*/(short)0, c, /*reuse_a=*/false, /*reuse_b=*/false);
}

// XOR bank swizzle (permutation within 16-element blocks): makes the
// stage-0 stride-16 group accesses conflict-free across the 64 LDS banks.
__device__ __forceinline__ int swz(int i) { return i ^ ((i >> 4) & 15); }

// LDS element index for group g (0..255), basis m (0..15), stage bit-offset s4
__device__ __forceinline__ int grp_idx(int g, int m, int s4) {
  int lo = g & ((1 << s4) - 1);
  int hi = g >> s4;
  return (hi << (s4 + 4)) | (m << s4) | lo;
}

// Entry (m,k) of Kron of four RX gates: RX diag = c, offdiag = -i*s
// => U[m][k] = (prod of c/s factors) * (-i)^popcount(m^k)
__device__ __forceinline__ void rx_kron_entry(const float* cj, const float* sj,
                                              int m, int k, float& ure, float& uim) {
  int x = (m ^ k) & 15;
  float mag = 1.0f;
#pragma unroll
  for (int j = 0; j < 4; ++j) mag *= ((x >> j) & 1) ? sj[j] : cj[j];
  int d = __popc(x) & 3;          // (-i)^d
  ure = (d == 0) ? mag : ((d == 2) ? -mag : 0.0f);
  uim = (d == 1) ? -mag : ((d == 3) ? mag : 0.0f);
}

// ---------------------------------------------------------------------------
// One pass: apply RX on 12 index bits to every amplitude.
// STRIDE==1    : tile = contiguous 4096 amps  (bits 0..11,  qubits 12..23)
// STRIDE==TILE : tile = 4096 amps stride 4096 (bits 12..23, qubits 0..11)
// FIRST        : read raw SoA input scaled by 1/||psi||
// APPLY_CZ     : fold CZ-chain diagonal into the store (last pass of a layer)
// ---------------------------------------------------------------------------
template <int STRIDE, bool FIRST, bool APPLY_CZ>
__global__ __launch_bounds__(TPB) void qsim_layer_pass(
    float2* __restrict__ psi,          // interleaved working state (d_out)
    const float* __restrict__ re_in,
    const float* __restrict__ im_in,
    const double* __restrict__ norm2_ws,
    const float* __restrict__ thetas,  // [NLAYER][NQ]
    int layer)
{
  constexpr int BITBASE = (STRIDE == 1) ? 0 : 12;
  __shared__ float2 tile[TILE];       // interleaved (re, im), XOR-swizzled

  const int tid = threadIdx.x;
  const long long tileBase =
      (STRIDE == 1) ? ((long long)blockIdx.x * TILE) : (long long)blockIdx.x;

  // ---- global -> LDS ----
  if (FIRST) {
    float scale = (float)(1.0 / sqrt(*norm2_ws));
#pragma unroll
    for (int it = 0; it < TILE / TPB; ++it) {
      int h = tid + it * TPB;
      long long gi = tileBase + (long long)h * STRIDE;
      tile[swz(h)] = make_float2(re_in[gi] * scale, im_in[gi] * scale);
    }
  } else {
#if HAS_ASYNC_LOAD
#pragma unroll
    for (int it = 0; it < TILE / TPB; ++it) {
      int h = tid + it * TPB;
      long long gi = tileBase + (long long)h * STRIDE;
      __builtin_amdgcn_global_load_async_to_lds_b64(
          (AS1 v2i*)(psi + gi), (AS3 v2i*)&tile[swz(h)], 0, 0);
    }
    WAIT_ASYNC0();
#else
#pragma unroll
    for (int it = 0; it < TILE / TPB; ++it) {
      int h = tid + it * TPB;
      long long gi = tileBase + (long long)h * STRIDE;
      tile[swz(h)] = psi[gi];
    }
#endif
  }
  __syncthreads();

  const int lane   = tid & 31;
  const int wave_u = __builtin_amdgcn_readfirstlane(tid >> 5);  // uniform SGPR
  const int mrow   = lane & 15;        // A-matrix row / D column selector
  const int khalf  = (lane >> 4) << 1; // lanes 16-31 hold K+2,K+3
  const int n      = lane & 15;

  // ---- three 4-qubit WMMA stages (fully unrolled; s4 compile-time) ----
#pragma unroll
  for (int st = 0; st < 3; ++st) {
    const int s4 = st * 4;

    // RX half-angles for this stage (index bit b <-> qubit 23-b).
    // theta/2 in [0, pi) -> hardware sin/cos is safe and accurate here.
    float cj[4], sj[4];
#pragma unroll
    for (int j = 0; j < 4; ++j) {
      int gb = BITBASE + s4 + j;
      float th = 0.5f * thetas[layer * NQ + (23 - gb)];
      cj[j] = __cosf(th);
      sj[j] = __sinf(th);
    }

    // A fragments: Ur, Ui, and -Ui (f32 WMMA has no A/B negate modifier)
    v2f ur[4], ui[4], un[4];
#pragma unroll
    for (int k = 0; k < 4; ++k) {
      int kc = 4 * k + khalf;
      float r0, i0, r1, i1;
      rx_kron_entry(cj, sj, mrow, kc,     r0, i0);
      rx_kron_entry(cj, sj, mrow, kc + 1, r1, i1);
      ur[k].x = r0;  ur[k].y = r1;
      ui[k].x = i0;  ui[k].y = i1;
      un[k].x = -i0; un[k].y = -i1;
    }

    // 16 batches of 16 groups; waves own disjoint groups -> barrier only
    // between stages. Loop bound/step uniform (SGPR) -> scalar branch, EXEC
    // stays all-ones around the WMMAs as the ISA requires.
    for (int b = wave_u; b < 16; b += NWAVE) {
      const int g = b * 16 + n;

      // B fragments of Pr / Pi (B 4x16: VGPR pair = K {4k,4k+1}/{4k+2,4k+3})
      v2f br[4], bi[4];
#pragma unroll
      for (int k = 0; k < 4; ++k) {
        int kc = 4 * k + khalf;
        float2 pA = tile[swz(grp_idx(g, kc,     s4))];
        float2 pB = tile[swz(grp_idx(g, kc + 1, s4))];
        br[k].x = pA.x; br[k].y = pB.x;
        bi[k].x = pA.y; bi[k].y = pB.y;
      }

      // Dr = Ur*Pr - Ui*Pi ; Di = Ur*Pi + Ui*Pr   (16 WMMAs)
      v8f dr = {0.f,0.f,0.f,0.f,0.f,0.f,0.f,0.f};
      v8f di = {0.f,0.f,0.f,0.f,0.f,0.f,0.f,0.f};
#pragma unroll
      for (int k = 0; k < 4; ++k) dr = wmma_f32_16x16x4(ur[k], br[k], dr);
#pragma unroll
      for (int k = 0; k < 4; ++k) dr = wmma_f32_16x16x4(un[k], bi[k], dr);
#pragma unroll
      for (int k = 0; k < 4; ++k) di = wmma_f32_16x16x4(ur[k], bi[k], di);
#pragma unroll
      for (int k = 0; k < 4; ++k) di = wmma_f32_16x16x4(ui[k], br[k], di);

      // D layout: VGPR v -> rows v (lanes 0-15) / v+8 (lanes 16-31), col = n
      const int rb = (lane >> 4) << 3;
#pragma unroll
      for (int v = 0; v < 8; ++v)
        tile[swz(grp_idx(g, rb + v, s4))] = make_float2(dr[v], di[v]);
    }
    __syncthreads();
  }

  // ---- LDS -> global (+ CZ diagonal on the layer's final pass) ----
  if (APPLY_CZ) {
#pragma unroll
    for (int it = 0; it < TILE / TPB; ++it) {
      int h = tid + it * TPB;
      long long gi = tileBase + (long long)h * STRIDE;
      float2 v = tile[swz(h)];
      unsigned int x   = (unsigned int)gi;
      unsigned int adj = x & (x >> 1) & 0x7FFFFFu;  // 23 adjacent-bit pairs
      if (__popc(adj) & 1) { v.x = -v.x; v.y = -v.y; }
      psi[gi] = v;
    }
  } else {
#if HAS_ASYNC_STORE
#pragma unroll
    for (int it = 0; it < TILE / TPB; ++it) {
      int h = tid + it * TPB;
      long long gi = tileBase + (long long)h * STRIDE;
      __builtin_amdgcn_global_store_async_from_lds_b64(
          (AS1 v2i*)(psi + gi), (AS3 v2i*)&tile[swz(h)], 0, 0);
    }
    WAIT_ASYNC0();
#else
#pragma unroll
    for (int it = 0; it < TILE / TPB; ++it) {
      int h = tid + it * TPB;
      long long gi = tileBase + (long long)h * STRIDE;
      psi[gi] = tile[swz(h)];
    }
#endif
  }
}

// ---------------------------------------------------------------------------
__global__ void zero_ws_kernel(double* ws) {
  ws[0] = 0.0;   // |psi|^2 of raw input
  ws[1] = 0.0;   // p0 after circuit
}

__global__ __launch_bounds__(256) void norm2_reduce(const float* __restrict__ re,
                                                    const float* __restrict__ im,
                                                    double* __restrict__ ws) {
  __shared__ double red[256];
  long long i0 = (long long)blockIdx.x * blockDim.x + threadIdx.x;
  long long gs = (long long)gridDim.x * blockDim.x;
  double acc = 0.0;
  for (long long i = i0; i < NSTATE; i += gs) {
    double r = re[i], m = im[i];
    acc += r * r + m * m;
  }
  red[threadIdx.x] = acc;
  __syncthreads();
  for (int s = 128; s > 0; s >>= 1) {
    if (threadIdx.x < s) red[threadIdx.x] += red[threadIdx.x + s];
    __syncthreads();
  }
  if (threadIdx.x == 0) atomicAdd(ws, red[0]);
}

__global__ __launch_bounds__(256) void p0_reduce(const float2* __restrict__ psi,
                                                 double* __restrict__ ws_p0) {
  __shared__ double red[256];
  long long i0 = (long long)blockIdx.x * blockDim.x + threadIdx.x;
  long long gs = (long long)gridDim.x * blockDim.x;
  double acc = 0.0;
  for (long long i = i0; i < (NSTATE / 2); i += gs) {  // index bit23 == 0
    float2 v = psi[i];
    acc += (double)v.x * v.x + (double)v.y * v.y;
  }
  red[threadIdx.x] = acc;
  __syncthreads();
  for (int s = 128; s > 0; s >>= 1) {
    if (threadIdx.x < s) red[threadIdx.x] += red[threadIdx.x + s];
    __syncthreads();
  }
  if (threadIdx.x == 0) atomicAdd(ws_p0, red[0]);
}

__global__ __launch_bounds__(256) void finalize_measure(float2* __restrict__ psi,
                                                        const float* __restrict__ u,
                                                        const double* __restrict__ ws_p0,
                                                        float* __restrict__ out_mp) {
  double p0 = *ws_p0;
  int   m   = (u[0] >= (float)p0) ? 1 : 0;
  double p  = m ? (1.0 - p0) : p0;
  float inv = (float)(1.0 / sqrt(p));
  long long i0 = (long long)blockIdx.x * blockDim.x + threadIdx.x;
  long long gs = (long long)gridDim.x * blockDim.x;
  for (long long i = i0; i < NSTATE; i += gs) {
    int keep = (int)((i >> 23) & 1) == m;
    float2 v = psi[i];
    psi[i] = keep ? make_float2(v.x * inv, v.y * inv) : make_float2(0.0f, 0.0f);
  }
  if (blockIdx.x == 0 && threadIdx.x == 0) {
    out_mp[0] = (float)m;   // measurement outcome
    out_mp[1] = (float)p;   // conditional probability
  }
}

// ---------------------------------------------------------------------------
extern "C" void kernel_launch(void* const* d_in, const int* in_sizes, int n_in,
                              void* d_out, int out_size, void* d_ws, size_t ws_size,
                              hipStream_t stream) {
  const float* psi_re = (const float*)d_in[0];
  const float* psi_im = (const float*)d_in[1];
  const float* thetas = (const float*)d_in[2];  // [4][24]
  const float* u      = (const float*)d_in[3];  // [1]

  float2* psi = (float2*)d_out;                 // working state, interleaved
  double* ws  = (double*)d_ws;                  // ws[0]=norm2, ws[1]=p0

  zero_ws_kernel<<<dim3(1), dim3(1), 0, stream>>>(ws);
  norm2_reduce<<<dim3(2048), dim3(256), 0, stream>>>(psi_re, psi_im, ws);

  const int nblocks = NSTATE / TILE;  // 4096 tiles

  // layer 0: pass1 reads+normalizes the raw input
  qsim_layer_pass<1, true, false><<<dim3(nblocks), dim3(TPB), 0, stream>>>(
      psi, psi_re, psi_im, ws, thetas, 0);
  qsim_layer_pass<TILE, false, true><<<dim3(nblocks), dim3(TPB), 0, stream>>>(
      psi, psi_re, psi_im, ws, thetas, 0);

  for (int l = 1; l < NLAYER; ++l) {
    qsim_layer_pass<1, false, false><<<dim3(nblocks), dim3(TPB), 0, stream>>>(
        psi, psi_re, psi_im, ws, thetas, l);
    qsim_layer_pass<TILE, false, true><<<dim3(nblocks), dim3(TPB), 0, stream>>>(
        psi, psi_re, psi_im, ws, thetas, l);
  }

  p0_reduce<<<dim3(2048), dim3(256), 0, stream>>>(psi, ws + 1);
  finalize_measure<<<dim3(4096), dim3(256), 0, stream>>>(
      psi, u, ws + 1, (float*)d_out + (size_t)NSTATE * 2);
}